// GNNModel_14362370638253
// MI455X (gfx1250) — compile-verified
//
#include <hip/hip_runtime.h>
#include <hip/hip_bf16.h>

typedef __attribute__((ext_vector_type(2))) float v2f;
typedef __attribute__((ext_vector_type(8))) float v8f;

// ---------------- degree / normalization ----------------

__global__ void deg_init_kernel(float* deg, int N) {
    int i = blockIdx.x * blockDim.x + threadIdx.x;
    if (i < N) deg[i] = 1.0f;  // self-loop contribution
}

__global__ void deg_accum_kernel(const long long* __restrict__ cols, float* deg, int E) {
    int i = blockIdx.x * blockDim.x + threadIdx.x;
    if (i < E) atomicAdd(&deg[(int)cols[i]], 1.0f);
}

__global__ void dinv_kernel(const float* __restrict__ deg, float* dinv, int N) {
    int i = blockIdx.x * blockDim.x + threadIdx.x;
    if (i < N) {
        float d = deg[i];
        dinv[i] = (d > 0.0f) ? rsqrtf(d) : 0.0f;
    }
}

// ---------------- GEMM: C[M,128] = A[M,128] @ W[128,128] via WMMA f32 16x16x4 ----------------

__launch_bounds__(256)
__global__ void gemm128_wmma_f32(const float* __restrict__ A,
                                 const float* __restrict__ W,
                                 float* __restrict__ C, int M) {
    __shared__ float Wl[128 * 128];
    // cooperative stage of W into LDS (64 KB)
    {
        const float4* src = (const float4*)W;
        float4* dst = (float4*)Wl;
        for (int i = threadIdx.x; i < (128 * 128) / 4; i += 256) dst[i] = src[i];
    }
    __syncthreads();

    const int m0   = blockIdx.x * 16;
    const int wave = threadIdx.x >> 5;   // 0..7 -> 16-col tile
    const int lane = threadIdx.x & 31;
    const int half = lane >> 4;          // 0: lanes 0-15, 1: lanes 16-31
    const int l16  = lane & 15;

    if (m0 + 16 <= M) {
        const int n0 = wave * 16;
        const float* Arow = A + (size_t)(m0 + l16) * 128;
        v8f acc = {};
        // A 16x4 layout: lanes0-15 hold M, VGPR0={K0|K2}, VGPR1={K1|K3}
        // B 4x16 layout (symmetric): lanes give N, VGPR0={K0|K2}, VGPR1={K1|K3}
        for (int k0 = 0; k0 < 128; k0 += 4) {
            const int ka = k0 + 2 * half;
            v2f a, b;
            a.x = Arow[ka];
            a.y = Arow[ka + 1];
            b.x = Wl[ka * 128 + n0 + l16];
            b.y = Wl[(ka + 1) * 128 + n0 + l16];
            acc = __builtin_amdgcn_wmma_f32_16x16x4_f32(
                false, a, false, b, (short)0, acc, false, false);
        }
        // C/D layout: VGPR j -> row m0 + j + 8*half, col n0 + l16
#pragma unroll
        for (int j = 0; j < 8; ++j)
            C[(size_t)(m0 + j + 8 * half) * 128 + n0 + l16] = acc[j];
    } else {
        // scalar tail (block-uniform branch; never taken when M % 16 == 0)
        const int rows = M - m0;
        for (int idx = threadIdx.x; idx < rows * 128; idx += 256) {
            const int r = idx >> 7, cc = idx & 127;
            float s = 0.0f;
            for (int k = 0; k < 128; ++k)
                s += A[(size_t)(m0 + r) * 128 + k] * Wl[k * 128 + cc];
            C[(size_t)(m0 + r) * 128 + cc] = s;
        }
    }
}

// ---------------- message passing ----------------

// acc[i, :] = dinv[i]^2 * xw[i, :]   (self-loop term)
__global__ void init_self_kernel(const float* __restrict__ xw,
                                 const float* __restrict__ dinv,
                                 float* __restrict__ acc, long long total) {
    long long i = (long long)blockIdx.x * blockDim.x + threadIdx.x;
    if (i < total) {
        const int node = (int)(i >> 7);
        const float d = dinv[node];
        acc[i] = d * d * xw[i];
    }
}

// one wave per edge: acc[col, :] += dinv[row]*dinv[col] * xw[row, :]
__global__ void scatter_edges_kernel(const long long* __restrict__ rows,
                                     const long long* __restrict__ cols,
                                     const float* __restrict__ dinv,
                                     const float* __restrict__ xw,
                                     float* __restrict__ acc, int E) {
    const int w = (int)(((long long)blockIdx.x * blockDim.x + threadIdx.x) >> 5);
    const int lane = threadIdx.x & 31;
    if (w >= E) return;
    const int r = (int)rows[w];
    const int c = (int)cols[w];
    const float norm = dinv[r] * dinv[c];
    const float4 v = ((const float4*)(xw + (size_t)r * 128))[lane];
    float* dst = acc + (size_t)c * 128 + lane * 4;
    atomicAdd(dst + 0, norm * v.x);
    atomicAdd(dst + 1, norm * v.y);
    atomicAdd(dst + 2, norm * v.z);
    atomicAdd(dst + 3, norm * v.w);
}

__global__ void bias_relu_kernel(float* __restrict__ acc,
                                 const float* __restrict__ b, long long total) {
    long long i = (long long)blockIdx.x * blockDim.x + threadIdx.x;
    if (i < total) {
        const float v = acc[i] + b[(int)(i & 127)];
        acc[i] = v > 0.0f ? v : 0.0f;
    }
}

// ---------------- pooling + readout ----------------

__global__ void zero_kernel(float* p, int n) {
    int i = blockIdx.x * blockDim.x + threadIdx.x;
    if (i < n) p[i] = 0.0f;
}

// one wave per node
__global__ void pool_accum_kernel(const float* __restrict__ h,
                                  const long long* __restrict__ batch,
                                  float* __restrict__ psum,
                                  float* __restrict__ pcnt, int N) {
    const int w = (int)(((long long)blockIdx.x * blockDim.x + threadIdx.x) >> 5);
    const int lane = threadIdx.x & 31;
    if (w >= N) return;
    const int g = (int)batch[w];
    const float4 v = ((const float4*)(h + (size_t)w * 128))[lane];
    float* dst = psum + (size_t)g * 128 + lane * 4;
    atomicAdd(dst + 0, v.x);
    atomicAdd(dst + 1, v.y);
    atomicAdd(dst + 2, v.z);
    atomicAdd(dst + 3, v.w);
    if (lane == 0) atomicAdd(pcnt + g, 1.0f);
}

// one wave (32 threads) per graph
__global__ void pool_final_kernel(const float* __restrict__ psum,
                                  const float* __restrict__ pcnt,
                                  const float* __restrict__ fc_w,
                                  const float* __restrict__ fc_b,
                                  float* __restrict__ out, int G) {
    const int g = blockIdx.x;
    const int lane = threadIdx.x;
    if (g >= G) return;
    const float inv = 1.0f / fmaxf(pcnt[g], 1.0f);
    float s = 0.0f;
    for (int f = lane; f < 128; f += 32)
        s += psum[(size_t)g * 128 + f] * inv * fc_w[f];
    for (int off = 16; off > 0; off >>= 1)
        s += __shfl_down(s, off, 32);
    if (lane == 0) out[g] = s + fc_b[0];
}

// ---------------- launcher ----------------

extern "C" void kernel_launch(void* const* d_in, const int* in_sizes, int n_in,
                              void* d_out, int out_size, void* d_ws, size_t ws_size,
                              hipStream_t stream) {
    const float*     x     = (const float*)d_in[0];
    const long long* ei    = (const long long*)d_in[1]; // [2, E] int64
    const long long* batch = (const long long*)d_in[2]; // [N] int64
    const float*     W1    = (const float*)d_in[3];
    const float*     b1    = (const float*)d_in[4];
    const float*     W2    = (const float*)d_in[5];
    const float*     b2    = (const float*)d_in[6];
    const float*     fc_w  = (const float*)d_in[7];
    const float*     fc_b  = (const float*)d_in[8];
    float*           out   = (float*)d_out;

    const int N = in_sizes[0] / 128;
    const int E = in_sizes[1] / 2;
    const int G = out_size;
    const long long totF = (long long)N * 128;

    const long long* rows = ei;
    const long long* cols = ei + E;

    // workspace layout (floats)
    float* ws   = (float*)d_ws;
    float* deg  = ws;                       // N
    float* dinv = deg + N;                  // N
    float* xw   = dinv + N;                 // N*128
    float* acc  = xw + totF;                // N*128  (h lives here in-place)
    float* psum = acc + totF;               // G*128
    float* pcnt = psum + (size_t)G * 128;   // G

    const int T = 256;
    const int gN  = (N + T - 1) / T;
    const int gE  = (E + T - 1) / T;
    const int gF  = (int)((totF + T - 1) / T);
    const int gEw = (E + (T / 32) - 1) / (T / 32);   // one wave per edge
    const int gNw = (N + (T / 32) - 1) / (T / 32);   // one wave per node
    const int gM  = (N + 15) / 16;

    // normalization
    deg_init_kernel<<<gN, T, 0, stream>>>(deg, N);
    deg_accum_kernel<<<gE, T, 0, stream>>>(cols, deg, E);
    dinv_kernel<<<gN, T, 0, stream>>>(deg, dinv, N);

    // ---- layer 1 ----
    gemm128_wmma_f32<<<gM, T, 0, stream>>>(x, W1, xw, N);
    init_self_kernel<<<gF, T, 0, stream>>>(xw, dinv, acc, totF);
    scatter_edges_kernel<<<gEw, T, 0, stream>>>(rows, cols, dinv, xw, acc, E);
    bias_relu_kernel<<<gF, T, 0, stream>>>(acc, b1, totF);   // h1 in acc

    // ---- layer 2 ----
    gemm128_wmma_f32<<<gM, T, 0, stream>>>(acc, W2, xw, N);  // xw2 = h1 @ W2
    init_self_kernel<<<gF, T, 0, stream>>>(xw, dinv, acc, totF);
    scatter_edges_kernel<<<gEw, T, 0, stream>>>(rows, cols, dinv, xw, acc, E);
    bias_relu_kernel<<<gF, T, 0, stream>>>(acc, b2, totF);   // h2 in acc

    // ---- pooling + readout ----
    zero_kernel<<<(G * 128 + G + T - 1) / T, T, 0, stream>>>(psum, G * 128 + G);
    pool_accum_kernel<<<gNw, T, 0, stream>>>(acc, batch, psum, pcnt, N);
    pool_final_kernel<<<G, 32, 0, stream>>>(psum, pcnt, fc_w, fc_b, out, G);
}